// realignment_layer_78357383348914
// MI455X (gfx1250) — compile-verified
//
#include <hip/hip_runtime.h>

typedef __attribute__((ext_vector_type(2))) float v2f;
typedef __attribute__((ext_vector_type(8))) float v8f;

// ---------------------------------------------------------------------------
// Kernel 1: per-batch inverse of delta_T.
// delta_T = [R t; 0 0 0 1], R from normalized quaternion.
// Exact inverse = [Rinv, -Rinv t; 0 1] with Rinv = general 3x3 inverse
// (adjugate / det) so we match jnp.linalg.inv numerics, not just R^T.
// Writes invT[b][16] row-major into workspace.
// ---------------------------------------------------------------------------
__global__ void build_inv_kernel(const float* __restrict__ q_in,
                                 const float* __restrict__ t_in,
                                 float* __restrict__ invT, int B) {
  int b = blockIdx.x * blockDim.x + threadIdx.x;
  if (b >= B) return;

  float qw = q_in[b * 4 + 0], qx = q_in[b * 4 + 1];
  float qy = q_in[b * 4 + 2], qz = q_in[b * 4 + 3];
  float inv_n = 1.0f / sqrtf(qw * qw + qx * qx + qy * qy + qz * qz);
  qw *= inv_n; qx *= inv_n; qy *= inv_n; qz *= inv_n;

  float r00 = 1.0f - 2.0f * (qy * qy + qz * qz);
  float r01 = 2.0f * (qx * qy - qz * qw);
  float r02 = 2.0f * (qx * qz + qy * qw);
  float r10 = 2.0f * (qx * qy + qz * qw);
  float r11 = 1.0f - 2.0f * (qx * qx + qz * qz);
  float r12 = 2.0f * (qy * qz - qx * qw);
  float r20 = 2.0f * (qx * qz - qy * qw);
  float r21 = 2.0f * (qy * qz + qx * qw);
  float r22 = 1.0f - 2.0f * (qx * qx + qy * qy);

  // general 3x3 inverse via adjugate
  float c00 = r11 * r22 - r12 * r21;
  float c01 = r12 * r20 - r10 * r22;
  float c02 = r10 * r21 - r11 * r20;
  float det = r00 * c00 + r01 * c01 + r02 * c02;
  float id = 1.0f / det;

  float i00 = c00 * id;
  float i01 = (r02 * r21 - r01 * r22) * id;
  float i02 = (r01 * r12 - r02 * r11) * id;
  float i10 = c01 * id;
  float i11 = (r00 * r22 - r02 * r20) * id;
  float i12 = (r02 * r10 - r00 * r12) * id;
  float i20 = c02 * id;
  float i21 = (r01 * r20 - r00 * r21) * id;
  float i22 = (r00 * r11 - r01 * r10) * id;

  float t0 = t_in[b * 3 + 0], t1 = t_in[b * 3 + 1], t2 = t_in[b * 3 + 2];
  float ti0 = -(i00 * t0 + i01 * t1 + i02 * t2);
  float ti1 = -(i10 * t0 + i11 * t1 + i12 * t2);
  float ti2 = -(i20 * t0 + i21 * t1 + i22 * t2);

  float* o = invT + b * 16;
  o[0] = i00; o[1] = i01; o[2]  = i02; o[3]  = ti0;
  o[4] = i10; o[5] = i11; o[6]  = i12; o[7]  = ti1;
  o[8] = i20; o[9] = i21; o[10] = i22; o[11] = ti2;
  o[12] = 0.0f; o[13] = 0.0f; o[14] = 0.0f; o[15] = 1.0f;
}

// ---------------------------------------------------------------------------
// Kernel 2: batch_T_pred[b] = invT[b] (4x4) @ T_mis[b] (4x4) via
// V_WMMA_F32_16X16X4_F32. One wave (32 threads) per batch.
//
// A operand (16x4 f32, 2 VGPRs): lanes 0-15 hold row M=lane, v0=K0, v1=K1;
//                                lanes 16-31 hold row M=lane-16, v0=K2, v1=K3.
// B operand (4x16 f32, 2 VGPRs): lanes 0-15 hold col N=lane, v0=K0, v1=K1;
//                                lanes 16-31 hold col N=lane-16, v0=K2, v1=K3.
// D (16x16 f32, 8 VGPRs): VGPR r, lanes 0-15 -> row r, col = lane.
// Rows/cols >= 4 are zero-padded.
// ---------------------------------------------------------------------------
__global__ void tpred_wmma_kernel(const float* __restrict__ invT,
                                  const float* __restrict__ Tmis,
                                  float* __restrict__ Tpred) {
  int b = blockIdx.x;
  int lane = threadIdx.x;   // 0..31, one full wave, EXEC all ones
  int r = lane & 15;
  int hi = lane >> 4;       // 0 -> K={0,1}, 1 -> K={2,3}

  const float* A  = invT + b * 16;
  const float* Bm = Tmis + b * 16;

  v2f a = {0.0f, 0.0f};
  v2f bb = {0.0f, 0.0f};
  if (r < 4) {
    a.x  = A[r * 4 + 2 * hi + 0];        // A[row r][K = 2*hi]
    a.y  = A[r * 4 + 2 * hi + 1];        // A[row r][K = 2*hi+1]
    bb.x = Bm[(2 * hi + 0) * 4 + r];     // B[K = 2*hi  ][col r]
    bb.y = Bm[(2 * hi + 1) * 4 + r];     // B[K = 2*hi+1][col r]
  }

  v8f c = {};
  v8f d = __builtin_amdgcn_wmma_f32_16x16x4_f32(
      /*neg_a=*/false, a, /*neg_b=*/false, bb,
      /*c_mod=*/(short)0, c, /*reuse_a=*/false, /*reuse_b=*/false);

  if (hi == 0 && r < 4) {
    Tpred[b * 16 + 0 * 4 + r] = d[0];
    Tpred[b * 16 + 1 * 4 + r] = d[1];
    Tpred[b * 16 + 2 * 4 + r] = d[2];
    Tpred[b * 16 + 3 * 4 + r] = d[3];
  }
}

// ---------------------------------------------------------------------------
// Kernel 3: streaming point transform. Memory-bound (0.75 flop/byte);
// each thread handles 4 points = 3 x float4 (48 contiguous bytes) so the
// wave issues full B128 loads/stores covering contiguous 1536-byte spans.
// Matrix/translation are block-uniform (scalarized by the compiler).
// ---------------------------------------------------------------------------
__global__ void pcd_transform_kernel(const float* __restrict__ pcd,
                                     const float* __restrict__ invT,
                                     float* __restrict__ out,
                                     int N, int chunksPerBatch) {
  int b = blockIdx.x / chunksPerBatch;
  int chunk = blockIdx.x % chunksPerBatch;

  const float* M = invT + b * 16;
  float m00 = M[0], m01 = M[1], m02 = M[2],  t0 = M[3];
  float m10 = M[4], m11 = M[5], m12 = M[6],  t1 = M[7];
  float m20 = M[8], m21 = M[9], m22 = M[10], t2 = M[11];

  int pib = chunk * (int)(blockDim.x * 4) + (int)threadIdx.x * 4;  // point idx in batch
  if (pib >= N) return;

  size_t base = ((size_t)b * (size_t)N + (size_t)pib) * 3;  // float offset

  if (pib + 4 <= N) {
    const float4* ip = (const float4*)(pcd + base);
    float4 f0 = ip[0];
    float4 f1 = ip[1];
    float4 f2 = ip[2];

    float x0 = f0.x, y0 = f0.y, z0 = f0.z;
    float x1 = f0.w, y1 = f1.x, z1 = f1.y;
    float x2 = f1.z, y2 = f1.w, z2 = f2.x;
    float x3 = f2.y, y3 = f2.z, z3 = f2.w;

    float ox0 = fmaf(m00, x0, fmaf(m01, y0, fmaf(m02, z0, t0)));
    float oy0 = fmaf(m10, x0, fmaf(m11, y0, fmaf(m12, z0, t1)));
    float oz0 = fmaf(m20, x0, fmaf(m21, y0, fmaf(m22, z0, t2)));
    float ox1 = fmaf(m00, x1, fmaf(m01, y1, fmaf(m02, z1, t0)));
    float oy1 = fmaf(m10, x1, fmaf(m11, y1, fmaf(m12, z1, t1)));
    float oz1 = fmaf(m20, x1, fmaf(m21, y1, fmaf(m22, z1, t2)));
    float ox2 = fmaf(m00, x2, fmaf(m01, y2, fmaf(m02, z2, t0)));
    float oy2 = fmaf(m10, x2, fmaf(m11, y2, fmaf(m12, z2, t1)));
    float oz2 = fmaf(m20, x2, fmaf(m21, y2, fmaf(m22, z2, t2)));
    float ox3 = fmaf(m00, x3, fmaf(m01, y3, fmaf(m02, z3, t0)));
    float oy3 = fmaf(m10, x3, fmaf(m11, y3, fmaf(m12, z3, t1)));
    float oz3 = fmaf(m20, x3, fmaf(m21, y3, fmaf(m22, z3, t2)));

    float4* op = (float4*)(out + base);
    op[0] = make_float4(ox0, oy0, oz0, ox1);
    op[1] = make_float4(oy1, oz1, ox2, oy2);
    op[2] = make_float4(oz2, ox3, oy3, oz3);
  } else {
    // tail (not hit for N % (blockDim*4) == 0, kept for generality)
    for (int i = pib; i < N; ++i) {
      size_t o = ((size_t)b * (size_t)N + (size_t)i) * 3;
      float x = pcd[o + 0], y = pcd[o + 1], z = pcd[o + 2];
      out[o + 0] = fmaf(m00, x, fmaf(m01, y, fmaf(m02, z, t0)));
      out[o + 1] = fmaf(m10, x, fmaf(m11, y, fmaf(m12, z, t1)));
      out[o + 2] = fmaf(m20, x, fmaf(m21, y, fmaf(m22, z, t2)));
    }
  }
}

extern "C" void kernel_launch(void* const* d_in, const int* in_sizes, int n_in,
                              void* d_out, int out_size, void* d_ws, size_t ws_size,
                              hipStream_t stream) {
  const float* pcd  = (const float*)d_in[0];  // (B, N, 3)
  const float* Tmis = (const float*)d_in[1];  // (B, 4, 4)
  const float* dq   = (const float*)d_in[2];  // (B, 4)
  const float* dt   = (const float*)d_in[3];  // (B, 3)

  int B = in_sizes[2] / 4;
  int N = in_sizes[0] / (B * 3);

  float* Tpred   = (float*)d_out;                      // first B*16 floats
  float* pcd_out = (float*)d_out + (size_t)B * 16;     // then B*N*3 floats
  float* invT    = (float*)d_ws;                       // B*16 floats scratch

  build_inv_kernel<<<dim3((B + 63) / 64), dim3(64), 0, stream>>>(dq, dt, invT, B);
  tpred_wmma_kernel<<<dim3(B), dim3(32), 0, stream>>>(invT, Tmis, Tpred);

  const int ptsPerBlock = 256 * 4;
  int chunks = (N + ptsPerBlock - 1) / ptsPerBlock;
  pcd_transform_kernel<<<dim3(B * chunks), dim3(256), 0, stream>>>(
      pcd, invT, pcd_out, N, chunks);
}